// STAR_42236708389197
// MI455X (gfx1250) — compile-verified
//
#include <hip/hip_runtime.h>
#include <hip/hip_bf16.h>
#include <math.h>

// ---------------- problem constants ----------------
#define BN   1024
#define VN   6890
#define NJ   24
#define NB   10
#define P3   (VN * 3)     // 20670, flattened (v,k)
#define FPAD 96           // pose_feat / posedirs K padded 93 -> 96
#define NBP  12           // betas / shapedirs K padded 10 -> 12

typedef float v2f __attribute__((ext_vector_type(2)));
typedef float v8f __attribute__((ext_vector_type(8)));

__constant__ int PARENTS_C[23] = {0,0,0,1,2,3,4,5,6,7,8,9,9,9,12,13,14,16,17,18,19,20,21};

// ======================================================================
// K-1: zero-padded copies of the K-contracted operands so the WMMA inner
// loops need no predication (K-tail must be true zeros; N-tail is handled
// by index clamping since those WMMA columns are never stored).
// ======================================================================
__global__ __launch_bounds__(256) void star_pad_kernel(
    const float* __restrict__ betas,      // B x 10
    const float* __restrict__ shapedirs,  // 20670 x 10
    const float* __restrict__ posedirs,   // 20670 x 93
    float* __restrict__ betas_pad,        // ws: B x 12
    float* __restrict__ shapedirs_pad,    // ws: 20670 x 12
    float* __restrict__ posedirs_pad)     // ws: 20670 x 96
{
    const int stride = gridDim.x * blockDim.x;
    const int tid0   = blockIdx.x * blockDim.x + threadIdx.x;
    for (int i = tid0; i < P3 * FPAD; i += stride) {
        const int p = i / FPAD, k = i - p * FPAD;
        posedirs_pad[i] = (k < 93) ? posedirs[(size_t)p * 93 + k] : 0.f;
    }
    for (int i = tid0; i < P3 * NBP; i += stride) {
        const int p = i / NBP, k = i - p * NBP;
        shapedirs_pad[i] = (k < NB) ? shapedirs[(size_t)p * NB + k] : 0.f;
    }
    for (int i = tid0; i < BN * NBP; i += stride) {
        const int b = i / NBP, k = i - b * NBP;
        betas_pad[i] = (k < NB) ? betas[(size_t)b * NB + k] : 0.f;
    }
}

// ======================================================================
// K0: per-joint rodrigues -> R (B,24,9), pose_feat (B,96 padded) in ws
// ======================================================================
__global__ __launch_bounds__(32) void star_pose_kernel(
    const float* __restrict__ pose,   // B x 72
    const float* __restrict__ betas,  // B x 10
    float* __restrict__ pose_feat,    // ws: B x 96
    float* __restrict__ Rws)          // ws: B x 24 x 9
{
    const int b = blockIdx.x;
    const int j = threadIdx.x;
    if (j == 0) {
        pose_feat[(size_t)b*FPAD + 92] = betas[(size_t)b*NB + 1];
        pose_feat[(size_t)b*FPAD + 93] = 0.f;
        pose_feat[(size_t)b*FPAD + 94] = 0.f;
        pose_feat[(size_t)b*FPAD + 95] = 0.f;
    }
    if (j < NJ) {
        const float tx = pose[(size_t)b*72 + j*3 + 0];
        const float ty = pose[(size_t)b*72 + j*3 + 1];
        const float tz = pose[(size_t)b*72 + j*3 + 2];
        const float ax = tx + 1e-8f, ay = ty + 1e-8f, az = tz + 1e-8f;
        const float angle = sqrtf(ax*ax + ay*ay + az*az);
        const float inv = 1.0f / angle;
        const float nx = tx*inv, ny = ty*inv, nz = tz*inv;
        const float half = 0.5f * angle;
        const float s = sinf(half), c = cosf(half);
        float qw = c, qx = s*nx, qy = s*ny, qz = s*nz;
        const float qn = rsqrtf(qw*qw + qx*qx + qy*qy + qz*qz);
        qw *= qn; qx *= qn; qy *= qn; qz *= qn;
        const float w2 = qw*qw, x2 = qx*qx, y2 = qy*qy, z2 = qz*qz;
        const float wx = qw*qx, wy = qw*qy, wz = qw*qz;
        const float xy = qx*qy, xz = qx*qz, yz = qy*qz;
        float* R = Rws + (size_t)b*NJ*9 + j*9;
        R[0] = w2 + x2 - y2 - z2; R[1] = 2.f*xy - 2.f*wz; R[2] = 2.f*wy + 2.f*xz;
        R[3] = 2.f*wz + 2.f*xy;   R[4] = w2 - x2 + y2 - z2; R[5] = 2.f*yz - 2.f*wx;
        R[6] = 2.f*xz - 2.f*wy;   R[7] = 2.f*wx + 2.f*yz;   R[8] = w2 - x2 - y2 + z2;
        if (j >= 1) {
            float* pf = pose_feat + (size_t)b*FPAD + (j-1)*4;
            pf[0] = s*nx; pf[1] = s*ny; pf[2] = s*nz; pf[3] = c - 1.0f;
        }
    }
}

// ======================================================================
// K1: fused WMMA GEMMs, one wave = 16(batch) x 32(p) tile, 2 accumulators.
//   v_shaped = shapedirs_pad . betas_pad + template      (K=12, 3x2 WMMA)
//   v_posed  = v_shaped + posedirs_pad . pose_feat       (K=96, 24x2 WMMA)
// All inner-loop loads are unconditional aligned b64; out-of-range p is
// index-clamped (garbage columns never stored).
// ======================================================================
__global__ __launch_bounds__(256) void star_gemm_kernel(
    const float* __restrict__ betas_pad,     // B x 12
    const float* __restrict__ v_template,    // 20670
    const float* __restrict__ shapedirs_pad, // 20670 x 12
    const float* __restrict__ posedirs_pad,  // 20670 x 96
    const float* __restrict__ pose_feat,     // B x 96
    float* __restrict__ out_vshaped,
    float* __restrict__ out_vposed)
{
    const int lane = threadIdx.x & 31;
    const int wv   = threadIdx.x >> 5;
    const int tile = blockIdx.x * 8 + wv;
    const int MT   = BN / 16;              // 64 batch tiles
    const int mt   = tile % MT;
    const int nt   = tile / MT;            // 0..645
    const int b0   = mt * 16;
    const int p0   = nt * 32;
    const int n    = lane & 15;
    const int hi   = lane >> 4;
    const int k2   = 2 * hi;               // lane-half K offset

    const int pA = min(p0 + n,      P3 - 1);
    const int pB = min(p0 + 16 + n, P3 - 1);

    const float* arow  = betas_pad + (size_t)(b0 + n) * NBP;
    const float* sdA   = shapedirs_pad + (size_t)pA * NBP;
    const float* sdB   = shapedirs_pad + (size_t)pB * NBP;

    // ---- shapedirs GEMM (K=12) ----
    v8f acc0 = {}, acc1 = {};
    #pragma unroll
    for (int k0 = 0; k0 < NBP; k0 += 4) {
        const v2f a   = *(const v2f*)(arow + k0 + k2);
        const v2f bm0 = *(const v2f*)(sdA  + k0 + k2);
        const v2f bm1 = *(const v2f*)(sdB  + k0 + k2);
        acc0 = __builtin_amdgcn_wmma_f32_16x16x4_f32(false, a, false, bm0,
                                                     (short)0, acc0, false, false);
        acc1 = __builtin_amdgcn_wmma_f32_16x16x4_f32(false, a, false, bm1,
                                                     (short)0, acc1, false, false);
    }
    const float vtA = v_template[pA];
    const float vtB = v_template[pB];
    #pragma unroll
    for (int r = 0; r < 8; ++r) { acc0[r] += vtA; acc1[r] += vtB; }

    const int brow = b0 + 8*hi;
    if (p0 + n < P3) {
        #pragma unroll
        for (int r = 0; r < 8; ++r)
            out_vshaped[(size_t)(brow + r)*P3 + p0 + n] = acc0[r];
    }
    if (p0 + 16 + n < P3) {
        #pragma unroll
        for (int r = 0; r < 8; ++r)
            out_vshaped[(size_t)(brow + r)*P3 + p0 + 16 + n] = acc1[r];
    }

    // ---- posedirs GEMM (K=96), accumulate on v_shaped ----
    const float* frow = pose_feat + (size_t)(b0 + n) * FPAD;
    const float* pdA  = posedirs_pad + (size_t)pA * FPAD;
    const float* pdB  = posedirs_pad + (size_t)pB * FPAD;
    #pragma unroll
    for (int k0 = 0; k0 < FPAD; k0 += 4) {
        const v2f a   = *(const v2f*)(frow + k0 + k2);
        const v2f bm0 = *(const v2f*)(pdA  + k0 + k2);
        const v2f bm1 = *(const v2f*)(pdB  + k0 + k2);
        acc0 = __builtin_amdgcn_wmma_f32_16x16x4_f32(false, a, false, bm0,
                                                     (short)0, acc0, false, false);
        acc1 = __builtin_amdgcn_wmma_f32_16x16x4_f32(false, a, false, bm1,
                                                     (short)0, acc1, false, false);
    }
    if (p0 + n < P3) {
        #pragma unroll
        for (int r = 0; r < 8; ++r)
            out_vposed[(size_t)(brow + r)*P3 + p0 + n] = acc0[r];
    }
    if (p0 + 16 + n < P3) {
        #pragma unroll
        for (int r = 0; r < 8; ++r)
            out_vposed[(size_t)(brow + r)*P3 + p0 + 16 + n] = acc1[r];
    }
}

// ======================================================================
// K2: J[b,j,k] = sum_v J_regressor[j,v] * v_shaped[b,v,k]
// ======================================================================
__global__ __launch_bounds__(256) void star_jreg_kernel(
    const float* __restrict__ vshaped,   // B x 20670
    const float* __restrict__ Jreg,      // 24 x 6890
    float* __restrict__ Jout)            // ws: B x 24 x 3
{
    const int b    = blockIdx.x;
    const int lane = threadIdx.x & 31;
    const int wv   = threadIdx.x >> 5;   // 0..7
    const float* vs = vshaped + (size_t)b * P3;
    float acc[3][3] = {{0.f}};
    for (int v = lane; v < VN; v += 32) {
        const float x = vs[v*3 + 0];
        const float y = vs[v*3 + 1];
        const float z = vs[v*3 + 2];
        #pragma unroll
        for (int jj = 0; jj < 3; ++jj) {
            const int j = wv + jj*8;
            const float w = Jreg[(size_t)j*VN + v];
            acc[jj][0] += w*x; acc[jj][1] += w*y; acc[jj][2] += w*z;
        }
    }
    #pragma unroll
    for (int jj = 0; jj < 3; ++jj) {
        #pragma unroll
        for (int k = 0; k < 3; ++k) {
            float s = acc[jj][k];
            for (int off = 16; off > 0; off >>= 1) s += __shfl_down(s, off, 32);
            if (lane == 0) Jout[(size_t)b*NJ*3 + (wv + jj*8)*3 + k] = s;
        }
    }
}

// ======================================================================
// K3: kinematic chain (the reference's "results" chain is identical to G,
// so J_transformed = G[:,:,:3,3] + trans).  Emits G' transposed + padded:
// GpT[b][entry 0..15][joint 0..23], entries 12..15 zeroed, so the skinning
// kernel's A-operand is an unconditional aligned b64 load.
// ======================================================================
__global__ __launch_bounds__(32) void star_kin_kernel(
    const float* __restrict__ Rws,    // B x 24 x 9
    const float* __restrict__ Jws,    // B x 24 x 3
    const float* __restrict__ trans,  // B x 3
    float* __restrict__ GpT,          // ws: B x 16 x 24
    float* __restrict__ outJt)        // B x 24 x 3
{
    __shared__ float G[NJ][16];
    __shared__ float L[NJ][16];
    __shared__ float Jl[NJ][3];
    const int b = blockIdx.x;
    const int t = threadIdx.x;

    if (t < NJ) {
        Jl[t][0] = Jws[(size_t)b*NJ*3 + t*3 + 0];
        Jl[t][1] = Jws[(size_t)b*NJ*3 + t*3 + 1];
        Jl[t][2] = Jws[(size_t)b*NJ*3 + t*3 + 2];
    }
    __syncthreads();
    if (t < NJ) {
        float jx = Jl[t][0], jy = Jl[t][1], jz = Jl[t][2];
        if (t > 0) {
            const int par = PARENTS_C[t-1];
            jx -= Jl[par][0]; jy -= Jl[par][1]; jz -= Jl[par][2];
        }
        const float* R = Rws + (size_t)b*NJ*9 + t*9;
        L[t][0]=R[0]; L[t][1]=R[1]; L[t][2]=R[2];  L[t][3]=jx;
        L[t][4]=R[3]; L[t][5]=R[4]; L[t][6]=R[5];  L[t][7]=jy;
        L[t][8]=R[6]; L[t][9]=R[7]; L[t][10]=R[8]; L[t][11]=jz;
        L[t][12]=0.f; L[t][13]=0.f; L[t][14]=0.f;  L[t][15]=1.f;
    }
    __syncthreads();
    if (t < 16) G[0][t] = L[0][t];
    __syncthreads();
    for (int i = 1; i < NJ; ++i) {
        const int par = PARENTS_C[i-1];
        if (t < 16) {
            const int r = t >> 2, c = t & 3;
            float s = 0.f;
            #pragma unroll
            for (int k = 0; k < 4; ++k) s += G[par][r*4 + k] * L[i][k*4 + c];
            G[i][t] = s;
        }
        __syncthreads();
    }
    if (t < NJ) {
        const float jx = Jl[t][0], jy = Jl[t][1], jz = Jl[t][2];
        float* g = GpT + (size_t)b * 16 * NJ;
        #pragma unroll
        for (int r = 0; r < 3; ++r) {
            const float a0 = G[t][r*4+0], a1 = G[t][r*4+1], a2 = G[t][r*4+2];
            g[(r*4+0)*NJ + t] = a0;
            g[(r*4+1)*NJ + t] = a1;
            g[(r*4+2)*NJ + t] = a2;
            g[(r*4+3)*NJ + t] = G[t][r*4+3] - (a0*jx + a1*jy + a2*jz);
        }
        #pragma unroll
        for (int e = 12; e < 16; ++e) g[e*NJ + t] = 0.f;   // pad entries
        #pragma unroll
        for (int k = 0; k < 3; ++k)
            outJt[(size_t)b*NJ*3 + t*3 + k] = G[t][k*4+3] + trans[(size_t)b*3 + k];
    }
}

// ======================================================================
// K4: skinning. Per (batch, 16-vertex tile): T = G'^T(16x24) x W^T(24x16)
// with M=matrix-entry, N=vertex, K=joint; 6x v_wmma_f32_16x16x4_f32.
// All loads unconditional (clamped vertex; padded G' entries).
// ======================================================================
__global__ __launch_bounds__(256) void star_skin_kernel(
    const float* __restrict__ GpT,      // ws: B x 16 x 24
    const float* __restrict__ weights,  // V x 24
    const float* __restrict__ vposed,   // B x 20670
    const float* __restrict__ trans,    // B x 3
    float* __restrict__ outv)           // B x 20670
{
    __shared__ float Tls[8][16][16];    // [wave][vertex][entry] = 8 KB
    const int lane = threadIdx.x & 31;
    const int wv   = threadIdx.x >> 5;
    const int b    = blockIdx.y;
    const int tile = blockIdx.x * 8 + wv;
    const int v0   = tile * 16;
    const int n    = lane & 15;
    const int hi   = lane >> 4;
    const int k2   = 2 * hi;

    const int vc = min(v0 + n, VN - 1);
    const float* grow = GpT + (size_t)b * 16 * NJ + n * NJ;   // A row (entry n)
    const float* wrow = weights + (size_t)vc * NJ;            // B col (vertex n)

    v8f acc = {};
    #pragma unroll
    for (int k0 = 0; k0 < NJ; k0 += 4) {
        const v2f a  = *(const v2f*)(grow + k0 + k2);
        const v2f bm = *(const v2f*)(wrow + k0 + k2);
        acc = __builtin_amdgcn_wmma_f32_16x16x4_f32(false, a, false, bm,
                                                    (short)0, acc, false, false);
    }
    // C: lane holds vertex n, VGPR r = entry r + 8*hi
    #pragma unroll
    for (int r = 0; r < 8; ++r) Tls[wv][n][r + 8*hi] = acc[r];
    __syncthreads();

    if (lane < 16) {
        const int v = v0 + lane;
        if (v < VN) {
            const float* T = Tls[wv][lane];
            const float x = vposed[(size_t)b*P3 + v*3 + 0];
            const float y = vposed[(size_t)b*P3 + v*3 + 1];
            const float z = vposed[(size_t)b*P3 + v*3 + 2];
            #pragma unroll
            for (int m = 0; m < 3; ++m) {
                const float o = T[m*4+0]*x + T[m*4+1]*y + T[m*4+2]*z + T[m*4+3]
                              + trans[(size_t)b*3 + m];
                outv[(size_t)b*P3 + v*3 + m] = o;
            }
        }
    }
}

// ======================================================================
extern "C" void kernel_launch(void* const* d_in, const int* in_sizes, int n_in,
                              void* d_out, int out_size, void* d_ws, size_t ws_size,
                              hipStream_t stream) {
    const float* pose       = (const float*)d_in[0];
    const float* betas      = (const float*)d_in[1];
    const float* trans      = (const float*)d_in[2];
    const float* v_template = (const float*)d_in[3];
    const float* shapedirs  = (const float*)d_in[4];
    const float* posedirs   = (const float*)d_in[5];
    const float* J_reg      = (const float*)d_in[6];
    const float* weights    = (const float*)d_in[7];

    float* out = (float*)d_out;
    const size_t bv3 = (size_t)BN * P3;
    float* out_v       = out;                 // (B,V,3)
    float* out_vposed  = out + bv3;           // (B,V,3)
    float* out_vshaped = out + 2*bv3;         // (B,V,3)
    float* out_Jt      = out + 3*bv3;         // (B,24,3)

    // workspace layout (floats; all sub-buffers 8B-aligned: even sizes)
    float* ws            = (float*)d_ws;
    float* betas_pad     = ws;                                    // B*12
    float* shapedirs_pad = betas_pad + (size_t)BN*NBP;            // P3*12
    float* posedirs_pad  = shapedirs_pad + (size_t)P3*NBP;        // P3*96
    float* pose_feat     = posedirs_pad + (size_t)P3*FPAD;        // B*96
    float* Rws           = pose_feat + (size_t)BN*FPAD;           // B*216
    float* Jws           = Rws + (size_t)BN*NJ*9;                 // B*72
    float* GpT           = Jws + (size_t)BN*NJ*3;                 // B*384

    star_pad_kernel<<<1024, 256, 0, stream>>>(
        betas, shapedirs, posedirs, betas_pad, shapedirs_pad, posedirs_pad);

    star_pose_kernel<<<BN, 32, 0, stream>>>(pose, betas, pose_feat, Rws);

    const int ntiles = (BN/16) * ((P3 + 31)/32);              // 64 * 646
    star_gemm_kernel<<<ntiles/8, 256, 0, stream>>>(
        betas_pad, v_template, shapedirs_pad, posedirs_pad, pose_feat,
        out_vshaped, out_vposed);

    star_jreg_kernel<<<BN, 256, 0, stream>>>(out_vshaped, J_reg, Jws);

    star_kin_kernel<<<BN, 32, 0, stream>>>(Rws, Jws, trans, GpT, out_Jt);

    const int vtiles = (VN + 15)/16;                          // 431
    star_skin_kernel<<<dim3((vtiles + 7)/8, BN), 256, 0, stream>>>(
        GpT, weights, out_vposed, trans, out_v);
}